// GraphSAGE_NC_15126874816626
// MI455X (gfx1250) — compile-verified
//
#include <hip/hip_runtime.h>

#define N_NODES 50000
#define N_EDGES 600000
#define D_FEAT  128
#define BN_EPS  1e-5f

typedef __attribute__((ext_vector_type(2))) float v2f;
typedef __attribute__((ext_vector_type(8))) float v8f;

// ---------------- degree / inverse-degree ----------------
__global__ void degree_kernel(const long long* __restrict__ dst,
                              float* __restrict__ cnt, int nE) {
    int e = blockIdx.x * blockDim.x + threadIdx.x;
    if (e < nE) atomicAdd(&cnt[(int)dst[e]], 1.0f);
}

__global__ void inv_kernel(const float* __restrict__ cnt,
                           float* __restrict__ inv, int n) {
    int i = blockIdx.x * blockDim.x + threadIdx.x;
    if (i < n) inv[i] = 1.0f / fmaxf(cnt[i], 1.0f);
}

// ---------------- edge scatter: one wave per edge ----------------
// lane l moves float4 (features 4l..4l+3): coalesced 512B gather + 4 f32 atomics
__global__ void scatter_kernel(const long long* __restrict__ src,
                               const long long* __restrict__ dst,
                               const float* __restrict__ feat,
                               float* __restrict__ agg, int nE) {
    int gid  = blockIdx.x * blockDim.x + threadIdx.x;
    int e    = gid >> 5;
    int lane = gid & 31;
    if (e >= nE) return;
    int s = (int)src[e];
    int d = (int)dst[e];
    const float4* fp = (const float4*)(feat + (size_t)s * D_FEAT);
    float4 v = fp[lane];
    float* ap = agg + (size_t)d * D_FEAT + lane * 4;
    atomicAdd(ap + 0, v.x);
    atomicAdd(ap + 1, v.y);
    atomicAdd(ap + 2, v.z);
    atomicAdd(ap + 3, v.w);
}

// ---------------- fused  (inv*agg)@Wl + b + x@Wr  [+ BN + ReLU] ----------------
// block = 128 threads = 4 waves. Block stages its 16 rows of agg (pre-scaled by
// 1/deg) and x in padded LDS; wave w owns col tile blockIdx.x*4 + w.
// DOUT is a template constant so all weight addresses in the fully-unrolled
// K-loop are immediate offsets; column masking is branch-free and only
// compiled for DOUT % 16 != 0 (the 47-col output layer).
template <int DOUT, bool APPLY_BN>
__global__ void __launch_bounds__(128)
sage_gemm_kernel(const float* __restrict__ agg, const float* __restrict__ xin,
                 const float* __restrict__ inv,
                 const float* __restrict__ Wl, const float* __restrict__ bias,
                 const float* __restrict__ Wr,
                 const float* __restrict__ gam, const float* __restrict__ bet,
                 const float* __restrict__ mu,  const float* __restrict__ var,
                 float* __restrict__ out)
{
    __shared__ float sA[16][132];   // mean-scaled aggregate tile (padded rows)
    __shared__ float sX[16][132];   // root-feature tile

    const int t  = threadIdx.x;
    const int m0 = blockIdx.y * 16;

    // cooperative fill: 2 x (16x128) floats, float4 per thread per pass
    #pragma unroll
    for (int i = 0; i < 4; ++i) {
        int q   = i * 128 + t;       // float4 slot in 16x(128/4) tile
        int row = q >> 5;
        int col = (q & 31) * 4;
        float  s  = inv[m0 + row];
        float4 va = *(const float4*)(agg + (size_t)(m0 + row) * D_FEAT + col);
        sA[row][col + 0] = va.x * s;
        sA[row][col + 1] = va.y * s;
        sA[row][col + 2] = va.z * s;
        sA[row][col + 3] = va.w * s;
        float4 vx = *(const float4*)(xin + (size_t)(m0 + row) * D_FEAT + col);
        sX[row][col + 0] = vx.x;
        sX[row][col + 1] = vx.y;
        sX[row][col + 2] = vx.z;
        sX[row][col + 3] = vx.w;
    }
    __syncthreads();

    constexpr int NTILE = (DOUT + 15) / 16;
    const int wave = t >> 5;
    const int lane = t & 31;
    const int jt   = blockIdx.x * 4 + wave;
    if (jt >= NTILE) return;                   // wave-uniform exit (EXEC stays full)
    const int j0 = jt * 16;

    const int m   = lane & 15;                 // A row index / B,C,D col index
    const int grp = lane >> 4;                 // K-pair selector
    const int col = j0 + m;

    // branch-free column handling (only the 47-col layer needs it)
    constexpr bool NEED_MASK = (DOUT % 16) != 0;
    const int   colc = NEED_MASK ? (col < DOUT ? col : DOUT - 1) : col;
    const float msk  = NEED_MASK ? (col < DOUT ? 1.0f : 0.0f) : 1.0f;

    // per-lane weight base pointers: lane reads rows {2*grp, 2*grp+1} + kk
    const float* pl0 = Wl + (size_t)(2 * grp) * DOUT + colc;
    const float* pl1 = pl0 + DOUT;
    const float* pr0 = Wr + (size_t)(2 * grp) * DOUT + colc;
    const float* pr1 = pr0 + DOUT;

    v8f acc = {};
    #pragma unroll
    for (int kk = 0; kk < D_FEAT; kk += 4) {
        const int k0 = kk + 2 * grp;
        v2f a0 = *(const v2f*)(&sA[m][k0]);    // ds_load_b64, imm offset
        v2f a1 = *(const v2f*)(&sX[m][k0]);
        v2f bl, br;
        bl.x = pl0[kk * DOUT] * msk;           // global_load_b32, imm offset
        bl.y = pl1[kk * DOUT] * msk;
        br.x = pr0[kk * DOUT] * msk;
        br.y = pr1[kk * DOUT] * msk;
        acc = __builtin_amdgcn_wmma_f32_16x16x4_f32(
                  false, a0, false, bl, (short)0, acc, false, false);
        acc = __builtin_amdgcn_wmma_f32_16x16x4_f32(
                  false, a1, false, br, (short)0, acc, false, false);
    }

    float bcol = bias[colc];
    float sc = 1.0f, sh = 0.0f;
    if (APPLY_BN) {
        sc = gam[colc] * rsqrtf(var[colc] + BN_EPS);
        sh = bet[colc] - mu[colc] * sc;
    }
    #pragma unroll
    for (int r = 0; r < 8; ++r) {
        int   row = m0 + r + 8 * grp;
        float val = acc[r] + bcol;
        if (APPLY_BN) val = fmaxf(val * sc + sh, 0.0f);
        if (!NEED_MASK || col < DOUT)
            out[(size_t)row * DOUT + col] = val;
    }
}

extern "C" void kernel_launch(void* const* d_in, const int* in_sizes, int n_in,
                              void* d_out, int out_size, void* d_ws, size_t ws_size,
                              hipStream_t stream) {
    const float*     x   = (const float*)d_in[0];
    const long long* ei  = (const long long*)d_in[1];
    const long long* src = ei;
    const long long* dst = ei + N_EDGES;
    const float* W1l = (const float*)d_in[2];
    const float* b1  = (const float*)d_in[3];
    const float* W1r = (const float*)d_in[4];
    const float* W2l = (const float*)d_in[5];
    const float* b2  = (const float*)d_in[6];
    const float* W2r = (const float*)d_in[7];
    const float* W3l = (const float*)d_in[8];
    const float* b3  = (const float*)d_in[9];
    const float* W3r = (const float*)d_in[10];
    const float* g1  = (const float*)d_in[11];
    const float* be1 = (const float*)d_in[12];
    const float* m1  = (const float*)d_in[13];
    const float* v1  = (const float*)d_in[14];
    const float* g2  = (const float*)d_in[15];
    const float* be2 = (const float*)d_in[16];
    const float* m2  = (const float*)d_in[17];
    const float* v2  = (const float*)d_in[18];

    float* ws  = (float*)d_ws;
    float* cnt = ws;                                   // 50000
    float* inv = ws + N_NODES;                         // 50000
    float* agg = ws + 2 * (size_t)N_NODES;             // 50000*128
    float* h1  = agg + (size_t)N_NODES * D_FEAT;       // 50000*128
    float* h2  = h1  + (size_t)N_NODES * D_FEAT;       // 50000*128
    float* outf = (float*)d_out;

    const size_t aggBytes = (size_t)N_NODES * D_FEAT * sizeof(float);
    const int scatterBlocks = (N_EDGES * 32 + 255) / 256;
    dim3 gblock(128);
    dim3 ggrid128(2, N_NODES / 16);   // 8 col tiles / 4 waves per block
    dim3 ggrid47(1, N_NODES / 16);    // 3 col tiles (wave 3 idles)

    // degrees (shared by all layers)
    hipMemsetAsync(cnt, 0, N_NODES * sizeof(float), stream);
    degree_kernel<<<(N_EDGES + 255) / 256, 256, 0, stream>>>(dst, cnt, N_EDGES);
    inv_kernel<<<(N_NODES + 255) / 256, 256, 0, stream>>>(cnt, inv, N_NODES);

    // layer 1
    hipMemsetAsync(agg, 0, aggBytes, stream);
    scatter_kernel<<<scatterBlocks, 256, 0, stream>>>(src, dst, x, agg, N_EDGES);
    sage_gemm_kernel<128, true><<<ggrid128, gblock, 0, stream>>>(
        agg, x, inv, W1l, b1, W1r, g1, be1, m1, v1, h1);

    // layer 2
    hipMemsetAsync(agg, 0, aggBytes, stream);
    scatter_kernel<<<scatterBlocks, 256, 0, stream>>>(src, dst, h1, agg, N_EDGES);
    sage_gemm_kernel<128, true><<<ggrid128, gblock, 0, stream>>>(
        agg, h1, inv, W2l, b2, W2r, g2, be2, m2, v2, h2);

    // layer 3 (no BN/ReLU, 47 output cols)
    hipMemsetAsync(agg, 0, aggBytes, stream);
    scatter_kernel<<<scatterBlocks, 256, 0, stream>>>(src, dst, h2, agg, N_EDGES);
    sage_gemm_kernel<47, false><<<ggrid47, gblock, 0, stream>>>(
        agg, h2, inv, W3l, b3, W3r, nullptr, nullptr, nullptr, nullptr, outf);
}